// GNNReID_83141976916647
// MI455X (gfx1250) — compile-verified
//
#include <hip/hip_runtime.h>
#include <hip/hip_bf16.h>

// GAT 2-layer forward for MI455X (gfx1250), wave32, WMMA f32 16x16x4.
//
// Layout (CDNA5 ISA 7.12.2, 32-bit operands, wave32):
//   A 16x4  (MxK): lane l, vgpr v -> (M = l%16, K = v + 2*(l/16))
//   B 4x16  (KxN): lane l, vgpr v -> (K = v + 2*(l/16), N = l%16)
//   C 16x16 (MxN): lane l, vgpr r -> (M = r + 8*(l/16), N = l%16)
//
// Each wave computes a 32x64 output tile: 2 A fragments x 4 B fragments ->
// 8 WMMA accumulators; every B-fragment load feeds two WMMAs.  All hot-loop
// addresses are stepped pointers so the backend uses immediate offsets
// (no per-iteration 64-bit multiplies).

typedef float v2f __attribute__((ext_vector_type(2)));
typedef float v8f __attribute__((ext_vector_type(8)));

#define N_NODES 2048
#define NEG_SLOPE 0.2f

#define WMMA_F32(A, B, C) \
  __builtin_amdgcn_wmma_f32_16x16x4_f32(false, (A), false, (B), (short)0, (C), false, false)

// ---------------------------------------------------------------------------
// Dense GEMM: C(MxN) = A(MxK) @ B(KxN), fp32 row-major.
// Wave -> 32x64 tile. Block = 128 threads = 4 waves. Grid: (N/64, M/32/4).
// ---------------------------------------------------------------------------
__global__ void wmma_gemm_f32(const float* __restrict__ A,
                              const float* __restrict__ B,
                              float* __restrict__ C,
                              int N, int K) {
  const int lane = threadIdx.x & 31;
  const int wave = threadIdx.x >> 5;
  const int lm = lane & 15;
  const int lh = lane >> 4;
  const int row0 = (blockIdx.y * (blockDim.x >> 5) + wave) * 32;
  const int col0 = blockIdx.x * 64;

  v8f c00 = {}, c01 = {}, c02 = {}, c03 = {};
  v8f c10 = {}, c11 = {}, c12 = {}, c13 = {};

  const float* aPtr0 = A + (size_t)(row0 + lm) * K + 2 * lh;   // A row, K-pair lo
  const float* aPtr1 = aPtr0 + (size_t)16 * K;                 // A row + 16
  const float* bPtr0 = B + (size_t)(2 * lh) * N + col0 + lm;   // B row k+2lh
  const float* bPtr1 = bPtr0 + N;                              // B row k+2lh+1
  const size_t bStep = (size_t)4 * N;

  for (int k = 0; k < K; k += 4) {
    v2f a0, a1;
    a0.x = aPtr0[0];  a0.y = aPtr0[1];
    a1.x = aPtr1[0];  a1.y = aPtr1[1];
    aPtr0 += 4;
    aPtr1 += 4;

    v2f b0, b1, b2, b3;
    b0.x = bPtr0[0];   b0.y = bPtr1[0];
    b1.x = bPtr0[16];  b1.y = bPtr1[16];
    b2.x = bPtr0[32];  b2.y = bPtr1[32];
    b3.x = bPtr0[48];  b3.y = bPtr1[48];
    bPtr0 += bStep;
    bPtr1 += bStep;

    c00 = WMMA_F32(a0, b0, c00);  c10 = WMMA_F32(a1, b0, c10);
    c01 = WMMA_F32(a0, b1, c01);  c11 = WMMA_F32(a1, b1, c11);
    c02 = WMMA_F32(a0, b2, c02);  c12 = WMMA_F32(a1, b2, c12);
    c03 = WMMA_F32(a0, b3, c03);  c13 = WMMA_F32(a1, b3, c13);
  }

#pragma unroll
  for (int r = 0; r < 8; ++r) {
    float* cRow0 = C + (size_t)(row0 + r + 8 * lh) * N + col0 + lm;
    cRow0[0]  = c00[r];
    cRow0[16] = c01[r];
    cRow0[32] = c02[r];
    cRow0[48] = c03[r];
    float* cRow1 = cRow0 + (size_t)16 * N;
    cRow1[0]  = c10[r];
    cRow1[16] = c11[r];
    cRow1[32] = c12[r];
    cRow1[48] = c13[r];
  }
}

// ---------------------------------------------------------------------------
// Attention scores: sl[i,h] = g[i,h,:].a[:HID], sr[i,h] = g[i,h,:].a[HID:]
// ---------------------------------------------------------------------------
__global__ void gat_scores(const float* __restrict__ g,
                           const float* __restrict__ a,
                           float* __restrict__ sl, float* __restrict__ sr,
                           int nPairs, int NH, int HID) {
  int p = blockIdx.x * blockDim.x + threadIdx.x;
  if (p >= nPairs) return;
  int i = p / NH, h = p % NH;
  const float* row = g + (size_t)i * NH * HID + (size_t)h * HID;
  float l = 0.f, r = 0.f;
  for (int f = 0; f < HID; ++f) {
    l = fmaf(row[f], a[f], l);
    r = fmaf(row[f], a[HID + f], r);
  }
  sl[p] = l;
  sr[p] = r;
}

// ---------------------------------------------------------------------------
// Softmax row stats: per (i, head): m = max_j e[i,j,h] (adj!=0 only) and
// s = sum_j exp(e - m).  One wave per node i, online (m,s) accumulators,
// shuffle-xor merge tree.  Sentinel -1e30 avoids inf-inf NaNs.
// ---------------------------------------------------------------------------
template <int NH>
__global__ void gat_stats(const int* __restrict__ adj,
                          const float* __restrict__ sl,
                          const float* __restrict__ sr,
                          float* __restrict__ mOut, float* __restrict__ sOut,
                          int n) {
  const int i = blockIdx.x;
  const int lane = threadIdx.x;
  float slv[NH], m[NH], s[NH];
#pragma unroll
  for (int h = 0; h < NH; ++h) {
    slv[h] = sl[i * NH + h];
    m[h] = -1e30f;
    s[h] = 0.f;
  }
  const int* adjRow = adj + (size_t)i * n;
  for (int j = lane; j < n; j += 32) {
    if (adjRow[j] != 0) {
#pragma unroll
      for (int h = 0; h < NH; ++h) {
        float t = slv[h] + sr[j * NH + h];
        float lr = t > 0.f ? t : NEG_SLOPE * t;
        float nm = fmaxf(m[h], lr);
        s[h] = s[h] * __expf(m[h] - nm) + __expf(lr - nm);
        m[h] = nm;
      }
    }
  }
#pragma unroll
  for (int off = 16; off > 0; off >>= 1) {
#pragma unroll
    for (int h = 0; h < NH; ++h) {
      float om = __shfl_xor(m[h], off, 32);
      float os = __shfl_xor(s[h], off, 32);
      float nm = fmaxf(m[h], om);
      s[h] = s[h] * __expf(m[h] - nm) + os * __expf(om - nm);
      m[h] = nm;
    }
  }
  if (lane == 0) {
#pragma unroll
    for (int h = 0; h < NH; ++h) {
      mOut[i * NH + h] = m[h];
      sOut[i * NH + h] = s[h];
    }
  }
}

// ---------------------------------------------------------------------------
// Fused attention aggregation as a WMMA GEMM; attention matrix built on the
// fly in the A-fragment layout (flash-attention style):
//   out[i, head*HID + f] = (1/ssum[i,h]) * sum_j p(i,j,h) * g[j, head*HID+f]
//   p = adj ? exp(leakyrelu(sl_i + sr_j) - m_i) : 0
// Wave -> 32x64 tile; sr loads and B fragments shared across both A frags.
// Epilogue: row-scale by 1/ssum, optional ELU (layer 1).
// ---------------------------------------------------------------------------
__global__ void gat_agg_wmma(const int* __restrict__ adj,
                             const float* __restrict__ sl,
                             const float* __restrict__ sr,
                             const float* __restrict__ m,
                             const float* __restrict__ ssum,
                             const float* __restrict__ g,
                             float* __restrict__ out,
                             int n, int NH, int HID, int doElu) {
  const int Ntot = NH * HID;
  const int lane = threadIdx.x & 31;
  const int wave = threadIdx.x >> 5;
  const int lm = lane & 15;
  const int lh = lane >> 4;
  const int row0 = (blockIdx.y * (blockDim.x >> 5) + wave) * 32;
  const int col0 = blockIdx.x * 64;
  const int head = col0 / HID;

  const int i0 = row0 + lm;                // A-fragment rows for this lane
  const int i1 = i0 + 16;
  const float sl_i0 = sl[i0 * NH + head];
  const float sl_i1 = sl[i1 * NH + head];
  const float m_i0 = m[i0 * NH + head];
  const float m_i1 = m[i1 * NH + head];

  // Stepped hot-loop pointers (this lane covers K = j0+2*lh, j0+2*lh+1)
  const int* adjPtr0 = adj + (size_t)i0 * n + 2 * lh;
  const int* adjPtr1 = adj + (size_t)i1 * n + 2 * lh;
  const float* srPtr = sr + (size_t)(2 * lh) * NH + head;
  const float* bPtr0 = g + (size_t)(2 * lh) * Ntot + col0 + lm;
  const float* bPtr1 = bPtr0 + Ntot;
  const size_t bStep = (size_t)4 * Ntot;
  const size_t srStep = (size_t)4 * NH;

  v8f c00 = {}, c01 = {}, c02 = {}, c03 = {};
  v8f c10 = {}, c11 = {}, c12 = {}, c13 = {};

  for (int j0 = 0; j0 < n; j0 += 4) {
    const float sr0 = srPtr[0];
    const float sr1 = srPtr[NH];
    srPtr += srStep;
    const int a00 = adjPtr0[0];
    const int a01 = adjPtr0[1];
    const int a10 = adjPtr1[0];
    const int a11 = adjPtr1[1];
    adjPtr0 += 4;
    adjPtr1 += 4;

    float t00 = sl_i0 + sr0, t01 = sl_i0 + sr1;
    float t10 = sl_i1 + sr0, t11 = sl_i1 + sr1;
    t00 = t00 > 0.f ? t00 : NEG_SLOPE * t00;
    t01 = t01 > 0.f ? t01 : NEG_SLOPE * t01;
    t10 = t10 > 0.f ? t10 : NEG_SLOPE * t10;
    t11 = t11 > 0.f ? t11 : NEG_SLOPE * t11;

    v2f av0, av1;
    av0.x = a00 ? __expf(t00 - m_i0) : 0.f;
    av0.y = a01 ? __expf(t01 - m_i0) : 0.f;
    av1.x = a10 ? __expf(t10 - m_i1) : 0.f;
    av1.y = a11 ? __expf(t11 - m_i1) : 0.f;

    v2f b0, b1, b2, b3;
    b0.x = bPtr0[0];   b0.y = bPtr1[0];
    b1.x = bPtr0[16];  b1.y = bPtr1[16];
    b2.x = bPtr0[32];  b2.y = bPtr1[32];
    b3.x = bPtr0[48];  b3.y = bPtr1[48];
    bPtr0 += bStep;
    bPtr1 += bStep;

    c00 = WMMA_F32(av0, b0, c00);  c10 = WMMA_F32(av1, b0, c10);
    c01 = WMMA_F32(av0, b1, c01);  c11 = WMMA_F32(av1, b1, c11);
    c02 = WMMA_F32(av0, b2, c02);  c12 = WMMA_F32(av1, b2, c12);
    c03 = WMMA_F32(av0, b3, c03);  c13 = WMMA_F32(av1, b3, c13);
  }

  float inv0[8], inv1[8];
#pragma unroll
  for (int r = 0; r < 8; ++r) {
    inv0[r] = 1.0f / ssum[(row0 + r + 8 * lh) * NH + head];
    inv1[r] = 1.0f / ssum[(row0 + 16 + r + 8 * lh) * NH + head];
  }

#pragma unroll
  for (int r = 0; r < 8; ++r) {
    float v00 = c00[r] * inv0[r], v01 = c01[r] * inv0[r];
    float v02 = c02[r] * inv0[r], v03 = c03[r] * inv0[r];
    float v10 = c10[r] * inv1[r], v11 = c11[r] * inv1[r];
    float v12 = c12[r] * inv1[r], v13 = c13[r] * inv1[r];
    if (doElu) {
      v00 = v00 > 0.f ? v00 : __expf(v00) - 1.0f;
      v01 = v01 > 0.f ? v01 : __expf(v01) - 1.0f;
      v02 = v02 > 0.f ? v02 : __expf(v02) - 1.0f;
      v03 = v03 > 0.f ? v03 : __expf(v03) - 1.0f;
      v10 = v10 > 0.f ? v10 : __expf(v10) - 1.0f;
      v11 = v11 > 0.f ? v11 : __expf(v11) - 1.0f;
      v12 = v12 > 0.f ? v12 : __expf(v12) - 1.0f;
      v13 = v13 > 0.f ? v13 : __expf(v13) - 1.0f;
    }
    float* oRow0 = out + (size_t)(row0 + r + 8 * lh) * Ntot + col0 + lm;
    oRow0[0]  = v00;
    oRow0[16] = v01;
    oRow0[32] = v02;
    oRow0[48] = v03;
    float* oRow1 = oRow0 + (size_t)16 * Ntot;
    oRow1[0]  = v10;
    oRow1[16] = v11;
    oRow1[32] = v12;
    oRow1[48] = v13;
  }
}

// ---------------------------------------------------------------------------
extern "C" void kernel_launch(void* const* d_in, const int* in_sizes, int n_in,
                              void* d_out, int out_size, void* d_ws, size_t ws_size,
                              hipStream_t stream) {
  const float* x   = (const float*)d_in[0];   // 2048 x 512
  const int*   adj = (const int*)d_in[1];     // 2048 x 2048
  const float* W1  = (const float*)d_in[2];   // 512 x 512
  const float* a1  = (const float*)d_in[3];   // 128
  const float* W2  = (const float*)d_in[4];   // 512 x 256
  const float* a2  = (const float*)d_in[5];   // 512
  float* out = (float*)d_out;                 // 2048 x 256

  float* ws  = (float*)d_ws;                  // ~10.3 MB of fp32 scratch
  float* g1  = ws;                            // 2048*512
  float* hb  = g1 + 2048 * 512;               // 2048*512 (post-ELU layer-1 out)
  float* g2  = hb + 2048 * 512;               // 2048*256
  float* sl1 = g2 + 2048 * 256;               // 2048*8
  float* sr1 = sl1 + 2048 * 8;
  float* m1  = sr1 + 2048 * 8;
  float* ss1 = m1 + 2048 * 8;
  float* sl2 = ss1 + 2048 * 8;                // 2048
  float* sr2 = sl2 + 2048;
  float* m2  = sr2 + 2048;
  float* ss2 = m2 + 2048;

  const dim3 blk(128);                        // 4 waves of 32
  const dim3 grid1(512 / 64, 2048 / 32 / 4);  // 32x64 tile per wave
  const dim3 grid2(256 / 64, 2048 / 32 / 4);

  // ----- Layer 1 (8 heads, hid 64, concat, ELU) -----
  wmma_gemm_f32<<<grid1, blk, 0, stream>>>(x, W1, g1, 512, 512);
  gat_scores<<<(2048 * 8 + 255) / 256, 256, 0, stream>>>(g1, a1, sl1, sr1, 2048 * 8, 8, 64);
  gat_stats<8><<<2048, 32, 0, stream>>>(adj, sl1, sr1, m1, ss1, N_NODES);
  gat_agg_wmma<<<grid1, blk, 0, stream>>>(adj, sl1, sr1, m1, ss1, g1, hb,
                                          N_NODES, 8, 64, /*elu=*/1);

  // ----- Layer 2 (1 head, hid 256, mean over 1 head == identity) -----
  wmma_gemm_f32<<<grid2, blk, 0, stream>>>(hb, W2, g2, 256, 512);
  gat_scores<<<(2048 + 255) / 256, 256, 0, stream>>>(g2, a2, sl2, sr2, 2048, 1, 256);
  gat_stats<1><<<2048, 32, 0, stream>>>(adj, sl2, sr2, m2, ss2, N_NODES);
  gat_agg_wmma<<<grid2, blk, 0, stream>>>(adj, sl2, sr2, m2, ss2, g2, out,
                                          N_NODES, 1, 256, /*elu=*/0);
}